// MultitaskCTM_34892314313363
// MI455X (gfx1250) — compile-verified
//
#include <hip/hip_runtime.h>
#include <math.h>

typedef float v2f __attribute__((ext_vector_type(2)));
typedef float v8f __attribute__((ext_vector_type(8)));

#define RSBN 0.9999950000374996f  // 1/sqrt(1+1e-5)

static __device__ __forceinline__ v8f wmma_f32_k4(v2f a, v2f b, v8f c) {
  // D = A(16x4,f32) * B(4x16,f32) + C(16x16,f32), wave32
  return __builtin_amdgcn_wmma_f32_16x16x4_f32(false, a, false, b, (short)0, c,
                                               false, false);
}

// ---------------------------------------------------------------------------
// Generic dense layer: C[M=128,N] = A[128,K] @ W[K,N] + bias[N]   (WMMA f32)
// grid = (ceil(N/64), 8), block = 128 (4 waves). K <= 768, K % 32 == 0.
// Out-of-range columns (N=1000 tail) are handled by CLAMPING the load index
// (garbage accumulators are simply never stored) -> no EXEC divergence in the
// hot loop, WMMA always runs with full EXEC.
// ---------------------------------------------------------------------------
#define GEMM_KMAX 768
__global__ void __launch_bounds__(128)
gemm_bias_wmma(const float* __restrict__ A, const float* __restrict__ W,
               const float* __restrict__ bias, float* __restrict__ C,
               int N, int K) {
  __shared__ float As[16 * (GEMM_KMAX + 1)];
  const int tid = threadIdx.x;
  const int wv = tid >> 5, lane = tid & 31;
  const int half = lane >> 4, l16 = lane & 15;
  const int m0 = blockIdx.y * 16;
  const int n0 = blockIdx.x * 64 + wv * 16;
  const int ldA = K + 1;
  for (int r = 0; r < 16; ++r)
    for (int k = tid; k < K; k += 128)
      As[r * ldA + k] = A[(m0 + r) * K + k];
  __syncthreads();

  v8f acc = {0.f, 0.f, 0.f, 0.f, 0.f, 0.f, 0.f, 0.f};
  const int col = n0 + l16;
  const bool cok = (col < N);
  const int colc = cok ? col : (N - 1);        // clamp, never masked
  const float* ap = &As[l16 * ldA + 2 * half];
  const float* wq = W + colc + (2 * half) * N; // rows kk+2*half, kk+2*half+1
#pragma unroll 8
  for (int kk = 0; kk < K; kk += 4) {
    v2f a; a.x = ap[kk]; a.y = ap[kk + 1];
    v2f b; b.x = wq[kk * N]; b.y = wq[kk * N + N];
    acc = wmma_f32_k4(a, b, acc);
  }
  if (cok) {
    const float bv = bias[col];
#pragma unroll
    for (int r = 0; r < 8; ++r) {
      const int row = m0 + r + 8 * half;
      C[row * N + col] = acc[r] + bv;
    }
  }
}

// ---------------------------------------------------------------------------
// LayerNorm + ReLU over rows of X[128,N]. grid = 128 rows, block = 256.
// ---------------------------------------------------------------------------
__global__ void __launch_bounds__(256)
ln_relu_k(const float* __restrict__ X, const float* __restrict__ g,
          const float* __restrict__ b, float* __restrict__ Y, int N) {
  __shared__ float s1[256], s2[256];
  const int row = blockIdx.x, tid = threadIdx.x;
  const float* xr = X + row * N;
  float s = 0.f, q = 0.f;
  for (int c = tid; c < N; c += 256) { float v = xr[c]; s += v; q += v * v; }
  s1[tid] = s; s2[tid] = q;
  __syncthreads();
  for (int st = 128; st > 0; st >>= 1) {
    if (tid < st) { s1[tid] += s1[tid + st]; s2[tid] += s2[tid + st]; }
    __syncthreads();
  }
  const float mu = s1[0] / N;
  const float var = s2[0] / N - mu * mu;
  const float rs = rsqrtf(var + 1e-5f);
  for (int c = tid; c < N; c += 256)
    Y[row * N + c] = fmaxf((xr[c] - mu) * rs * g[c] + b[c], 0.f);
}

// ---------------------------------------------------------------------------
// CNN front end (fused conv+BN+ReLU, pools fused into conv2/conv3)
// ---------------------------------------------------------------------------
__global__ void __launch_bounds__(256)
conv1_k(const float* __restrict__ X, const float* __restrict__ W,
        const float* __restrict__ cb, const float* __restrict__ g,
        const float* __restrict__ bb, float* __restrict__ O) {
  const int idx = blockIdx.x * 256 + threadIdx.x;
  if (idx >= 128 * 16 * 64 * 64) return;
  const int x = idx & 63, y = (idx >> 6) & 63, oc = (idx >> 12) & 15,
            b = idx >> 16;
  float s = cb[oc];
  for (int ic = 0; ic < 3; ++ic) {
    const float* xp = X + ((b * 3 + ic) * 64) * 64;
    const float* wp = W + (oc * 3 + ic) * 9;
    for (int ky = 0; ky < 3; ++ky) {
      const int iy = y + ky - 1;
      if (iy < 0 || iy > 63) continue;
      for (int kx = 0; kx < 3; ++kx) {
        const int ix = x + kx - 1;
        if (ix < 0 || ix > 63) continue;
        s += xp[iy * 64 + ix] * wp[ky * 3 + kx];
      }
    }
  }
  O[idx] = fmaxf(s * RSBN * g[oc] + bb[oc], 0.f);
}

__global__ void __launch_bounds__(256)
conv2pool_k(const float* __restrict__ X, const float* __restrict__ W,
            const float* __restrict__ cb, const float* __restrict__ g,
            const float* __restrict__ bb, float* __restrict__ O) {
  const int idx = blockIdx.x * 256 + threadIdx.x;
  if (idx >= 128 * 32 * 32 * 32) return;
  const int px = idx & 31, py = (idx >> 5) & 31, oc = (idx >> 10) & 31,
            b = idx >> 15;
  const float scale = RSBN * g[oc], beta = bb[oc], cbo = cb[oc];
  float best = -3.4e38f;
  for (int sy = 0; sy < 2; ++sy)
    for (int sx = 0; sx < 2; ++sx) {
      const int y = py * 2 + sy, x = px * 2 + sx;
      float s = cbo;
      for (int ic = 0; ic < 16; ++ic) {
        const float* xp = X + ((b * 16 + ic) * 64) * 64;
        const float* wp = W + (oc * 16 + ic) * 9;
        for (int ky = 0; ky < 3; ++ky) {
          const int iy = y + ky - 1;
          if (iy < 0 || iy > 63) continue;
          for (int kx = 0; kx < 3; ++kx) {
            const int ix = x + kx - 1;
            if (ix < 0 || ix > 63) continue;
            s += xp[iy * 64 + ix] * wp[ky * 3 + kx];
          }
        }
      }
      best = fmaxf(best, s * scale + beta);
    }
  O[idx] = fmaxf(best, 0.f);
}

__global__ void __launch_bounds__(256)
conv3pool_k(const float* __restrict__ X, const float* __restrict__ W,
            const float* __restrict__ cb, const float* __restrict__ g,
            const float* __restrict__ bb, float* __restrict__ O) {
  const int idx = blockIdx.x * 256 + threadIdx.x;
  if (idx >= 128 * 64 * 16 * 16) return;
  const int px = idx & 15, py = (idx >> 4) & 15, oc = (idx >> 8) & 63,
            b = idx >> 14;
  const float scale = RSBN * g[oc], beta = bb[oc], cbo = cb[oc];
  float best = -3.4e38f;
  for (int sy = 0; sy < 2; ++sy)
    for (int sx = 0; sx < 2; ++sx) {
      const int y = py * 2 + sy, x = px * 2 + sx;
      float s = cbo;
      for (int ic = 0; ic < 32; ++ic) {
        const float* xp = X + ((b * 32 + ic) * 32) * 32;
        const float* wp = W + (oc * 32 + ic) * 9;
        for (int ky = 0; ky < 3; ++ky) {
          const int iy = y + ky - 1;
          if (iy < 0 || iy > 31) continue;
          for (int kx = 0; kx < 3; ++kx) {
            const int ix = x + kx - 1;
            if (ix < 0 || ix > 31) continue;
            s += xp[iy * 32 + ix] * wp[ky * 3 + kx];
          }
        }
      }
      best = fmaxf(best, s * scale + beta);
    }
  O[idx] = fmaxf(best, 0.f);
}

__global__ void __launch_bounds__(256)
conv4_k(const float* __restrict__ X, const float* __restrict__ W,
        const float* __restrict__ cb, const float* __restrict__ g,
        const float* __restrict__ bb, float* __restrict__ O) {
  const int idx = blockIdx.x * 256 + threadIdx.x;
  if (idx >= 128 * 64 * 16 * 16) return;
  const int x = idx & 15, y = (idx >> 4) & 15, oc = (idx >> 8) & 63,
            b = idx >> 14;
  float s = cb[oc];
  for (int ic = 0; ic < 64; ++ic) {
    const float* xp = X + ((b * 64 + ic) * 16) * 16;
    const float* wp = W + (oc * 64 + ic) * 9;
    for (int ky = 0; ky < 3; ++ky) {
      const int iy = y + ky - 1;
      if (iy < 0 || iy > 15) continue;
      for (int kx = 0; kx < 3; ++kx) {
        const int ix = x + kx - 1;
        if (ix < 0 || ix > 15) continue;
        s += xp[iy * 16 + ix] * wp[ky * 3 + kx];
      }
    }
  }
  O[idx] = fmaxf(s * RSBN * g[oc] + bb[oc], 0.f);
}

// (b,64,2,8,2,8).mean(3,5) -> feat[b, c*4 + i*2 + j]
__global__ void __launch_bounds__(256)
avg8_k(const float* __restrict__ A, float* __restrict__ F) {
  const int idx = blockIdx.x * 256 + threadIdx.x;
  if (idx >= 128 * 256) return;
  const int j = idx & 1, i = (idx >> 1) & 1, c = (idx >> 2) & 63, b = idx >> 8;
  const float* base = A + ((b * 64 + c) * 16 + i * 8) * 16 + j * 8;
  float s = 0.f;
  for (int y = 0; y < 8; ++y)
    for (int x = 0; x < 8; ++x) s += base[y * 16 + x];
  F[idx] = s * (1.f / 64.f);
}

// ---------------------------------------------------------------------------
// Tick-loop helpers
// ---------------------------------------------------------------------------
__global__ void __launch_bounds__(256)
s_init_k(const float* __restrict__ enc, float* __restrict__ S) {
  const int idx = blockIdx.x * 256 + threadIdx.x;  // 128*256
  if (idx >= 128 * 256) return;
  const int b = idx >> 8, c = idx & 255;
  S[b * 768 + c] = enc[idx];
}

__global__ void __launch_bounds__(256)
s_tail_k(const float* __restrict__ initial_post,
         const float* __restrict__ post_mem, float* __restrict__ S, int t) {
  const int idx = blockIdx.x * 256 + threadIdx.x;  // 128*512
  if (idx >= 128 * 512) return;
  const int b = idx >> 9, nn = idx & 511;
  const float v = (t == 0) ? initial_post[nn * 32 + 31]
                           : post_mem[(b * 512 + nn) * 32 + (t - 1)];
  S[b * 768 + 256 + nn] = v;
}

// preT[t][n][b] = pre[t][b][n]  (for coalesced A-gather in nlm kernel)
__global__ void __launch_bounds__(256)
pre_tr_k(const float* __restrict__ pre_t, float* __restrict__ preT_t) {
  const int idx = blockIdx.x * 256 + threadIdx.x;  // 128*512
  if (idx >= 128 * 512) return;
  const int b = idx >> 9, n = idx & 511;
  preT_t[n * 128 + b] = pre_t[b * 512 + n];
}

// ---------------------------------------------------------------------------
// Per-neuron memory MLP: h2 = LN(pre[b,n,:] @ nlm1_w[n] + nlm1_b[n]); relu;
// new_post[b,n] = h2 . nlm2_w[n] + nlm2_b[n].
// One workgroup per neuron (512), 8 waves, WMMA f32 16x16x4, K=32, N=128.
// ---------------------------------------------------------------------------
__global__ void __launch_bounds__(256)
nlm_tick_wmma(const float* __restrict__ initial_pre,
              const float* __restrict__ preT,   // [tick][n][b]
              const float* __restrict__ nlm1_w, const float* __restrict__ nlm1_b,
              const float* __restrict__ lng, const float* __restrict__ lnb,
              const float* __restrict__ w2, const float* __restrict__ b2,
              float* __restrict__ post_mem, int t) {
  __shared__ float As[128 * 33];   // A: 128 batches x 32 mem (padded)
  __shared__ float Bs[32 * 128];   // B: nlm1_w[n]
  const int n = blockIdx.x;
  const int tid = threadIdx.x;

  for (int idx = tid; idx < 128 * 32; idx += 256) {
    const int m = idx >> 7, b = idx & 127;  // b fastest -> coalesced preT read
    const float v = (m < 31 - t)
                        ? initial_pre[n * 32 + m + t + 1]
                        : preT[(m + t - 31) * (128 * 512) + n * 128 + b];
    As[b * 33 + m] = v;
  }
  for (int idx = tid; idx < 32 * 128; idx += 256)
    Bs[idx] = nlm1_w[n * 4096 + idx];
  __syncthreads();

  const int wv = tid >> 5, lane = tid & 31;
  const int half = lane >> 4, l16 = lane & 15;
  const int m0 = wv * 16;

  const v8f zero = {0.f, 0.f, 0.f, 0.f, 0.f, 0.f, 0.f, 0.f};
  v8f acc[8];
#pragma unroll
  for (int tt = 0; tt < 8; ++tt) acc[tt] = zero;

  const float* ap = &As[(m0 + l16) * 33 + 2 * half];
#pragma unroll
  for (int kk = 0; kk < 32; kk += 4) {
    v2f a; a.x = ap[kk]; a.y = ap[kk + 1];
    const int r0 = kk + 2 * half;
#pragma unroll
    for (int tt = 0; tt < 8; ++tt) {
      v2f b;
      b.x = Bs[r0 * 128 + tt * 16 + l16];
      b.y = Bs[(r0 + 1) * 128 + tt * 16 + l16];
      acc[tt] = wmma_f32_k4(a, b, acc[tt]);
    }
  }

  // per-column params + h2 bias
  float g8[8], bb8[8], w28[8];
#pragma unroll
  for (int tt = 0; tt < 8; ++tt) {
    const int col = n * 128 + tt * 16 + l16;
    const float b1 = nlm1_b[col];
    g8[tt] = lng[col]; bb8[tt] = lnb[col]; w28[tt] = w2[col];
#pragma unroll
    for (int r = 0; r < 8; ++r) acc[tt][r] += b1;
  }
  const float bias2 = b2[n];

  // register-resident LN over 128 cols (8 tiles x 16 half-lanes), relu, dot
#pragma unroll
  for (int r = 0; r < 8; ++r) {
    float s = 0.f, q = 0.f;
#pragma unroll
    for (int tt = 0; tt < 8; ++tt) { float v = acc[tt][r]; s += v; q += v * v; }
#pragma unroll
    for (int m = 1; m < 16; m <<= 1) {
      s += __shfl_xor(s, m, 32);
      q += __shfl_xor(q, m, 32);
    }
    const float mu = s * (1.f / 128.f);
    const float var = q * (1.f / 128.f) - mu * mu;
    const float rs = rsqrtf(var + 1e-5f);
    float d = 0.f;
#pragma unroll
    for (int tt = 0; tt < 8; ++tt) {
      float v = (acc[tt][r] - mu) * rs * g8[tt] + bb8[tt];
      v = fmaxf(v, 0.f);
      d += v * w28[tt];
    }
#pragma unroll
    for (int m = 1; m < 16; m <<= 1) d += __shfl_xor(d, m, 32);
    if (l16 == 0) {
      const int row = m0 + r + 8 * half;
      post_mem[(row * 512 + n) * 32 + t] = d + bias2;
    }
  }
}

// ---------------------------------------------------------------------------
// sync[b,p] = sum_m Zi*R*Zj / (sqrt(sum R)+1e-8), R = exp(-decay*(31-m))
// ---------------------------------------------------------------------------
__global__ void __launch_bounds__(256)
sync_k(const float* __restrict__ post_mem, const int* __restrict__ pairs,
       const float* __restrict__ decay, float* __restrict__ S) {
  const int idx = blockIdx.x * 256 + threadIdx.x;  // 128*256
  if (idx >= 128 * 256) return;
  const int p = idx & 255, b = idx >> 8;
  const int i = pairs[p * 2], j = pairs[p * 2 + 1];
  const float dec = decay[p];
  const float* zi = post_mem + (b * 512 + i) * 32;
  const float* zj = post_mem + (b * 512 + j) * 32;
  float rsum = 0.f, acc = 0.f;
#pragma unroll
  for (int m = 0; m < 32; ++m) {
    const float R = expf(-dec * (float)(31 - m));
    rsum += R;
    acc += zi[m] * R * zj[m];
  }
  S[b * 256 + p] = acc / (sqrtf(rsum) + 1e-8f);
}

// comb = [values | sync]  (softmax over singleton axis == 1 => attended==values)
__global__ void __launch_bounds__(256)
comb_k(const float* __restrict__ vals, const float* __restrict__ sync,
       float* __restrict__ C) {
  const int idx = blockIdx.x * 256 + threadIdx.x;  // 128*320
  if (idx >= 128 * 320) return;
  const int b = idx / 320, c = idx % 320;
  C[idx] = (c < 64) ? vals[b * 64 + c] : sync[b * 256 + (c - 64)];
}

// ---------------------------------------------------------------------------
extern "C" void kernel_launch(void* const* d_in, const int* in_sizes, int n_in,
                              void* d_out, int out_size, void* d_ws,
                              size_t ws_size, hipStream_t stream) {
  const float* x        = (const float*)d_in[0];
  const float* conv1_w  = (const float*)d_in[1];
  const float* conv1_b  = (const float*)d_in[2];
  const float* bn1_g    = (const float*)d_in[3];
  const float* bn1_b    = (const float*)d_in[4];
  const float* conv2_w  = (const float*)d_in[5];
  const float* conv2_b  = (const float*)d_in[6];
  const float* bn2_g    = (const float*)d_in[7];
  const float* bn2_b    = (const float*)d_in[8];
  const float* conv3_w  = (const float*)d_in[9];
  const float* conv3_b  = (const float*)d_in[10];
  const float* bn3_g    = (const float*)d_in[11];
  const float* bn3_b    = (const float*)d_in[12];
  const float* conv4_w  = (const float*)d_in[13];
  const float* conv4_b  = (const float*)d_in[14];
  const float* bn4_g    = (const float*)d_in[15];
  const float* bn4_b    = (const float*)d_in[16];
  const float* fc1_w    = (const float*)d_in[17];
  const float* fc1_b    = (const float*)d_in[18];
  const float* fcln_g   = (const float*)d_in[19];
  const float* fcln_b   = (const float*)d_in[20];
  const float* fc2_w    = (const float*)d_in[21];
  const float* fc2_b    = (const float*)d_in[22];
  // d_in[23]=k_w, d_in[24]=k_b : unused (softmax over singleton == 1)
  const float* v_w      = (const float*)d_in[25];
  const float* v_b      = (const float*)d_in[26];
  const float* init_pre = (const float*)d_in[27];
  const float* init_post= (const float*)d_in[28];
  const float* syn1_w   = (const float*)d_in[29];
  const float* syn1_b   = (const float*)d_in[30];
  const float* synln_g  = (const float*)d_in[31];
  const float* synln_b  = (const float*)d_in[32];
  const float* syn2_w   = (const float*)d_in[33];
  const float* syn2_b   = (const float*)d_in[34];
  const float* nlm1_w   = (const float*)d_in[35];
  const float* nlm1_b   = (const float*)d_in[36];
  const float* nlmln_g  = (const float*)d_in[37];
  const float* nlmln_b  = (const float*)d_in[38];
  const float* nlm2_w   = (const float*)d_in[39];
  const float* nlm2_b   = (const float*)d_in[40];
  const float* decay    = (const float*)d_in[41];
  // d_in[42]=q_w, d_in[43]=q_b : unused (scores never affect attw==1)
  const float* r1_w     = (const float*)d_in[44];
  const float* r1_b     = (const float*)d_in[45];
  const float* rln1_g   = (const float*)d_in[46];
  const float* rln1_b   = (const float*)d_in[47];
  const float* r2_w     = (const float*)d_in[48];
  const float* r2_b     = (const float*)d_in[49];
  const float* rln2_g   = (const float*)d_in[50];
  const float* rln2_b   = (const float*)d_in[51];
  const float* r3_w     = (const float*)d_in[52];
  const float* r3_b     = (const float*)d_in[53];
  const int*   pairs    = (const int*)d_in[54];
  float* out = (float*)d_out;

  float* W = (float*)d_ws;
  size_t off = 0;
  auto alloc = [&](size_t nElems) { float* p = W + off; off += nElems; return p; };

  float* a1      = alloc(128ull * 16 * 64 * 64);  // conv1 out
  float* a2      = alloc(128ull * 32 * 32 * 32);  // conv2+pool out
  float* a3      = alloc(128ull * 64 * 16 * 16);  // conv3+pool out
  float* a4      = alloc(128ull * 64 * 16 * 16);  // conv4 out
  float* feat    = alloc(128ull * 256);
  float* fcm     = alloc(128ull * 128);
  float* fcm2    = alloc(128ull * 128);
  float* enc     = alloc(128ull * 256);
  float* vals    = alloc(128ull * 64);
  float* S       = alloc(128ull * 768);
  float* h1      = alloc(128ull * 256);
  float* h1n     = alloc(128ull * 256);
  float* pre_mem = alloc(32ull * 128 * 512);   // [t][b][n]
  float* preT    = alloc(32ull * 128 * 512);   // [t][n][b]
  float* postm   = alloc(128ull * 512 * 32);   // [b][n][t]
  float* syncb   = alloc(128ull * 256);
  float* comb    = alloc(128ull * 320);
  float* hr1     = alloc(128ull * 512);
  float* hr1n    = alloc(128ull * 512);
  float* hr2     = alloc(128ull * 64);
  float* hr2n    = alloc(128ull * 64);
  (void)ws_size; (void)in_sizes; (void)n_in; (void)out_size;

  // --- CNN front end ---
  conv1_k<<<(128 * 16 * 64 * 64 + 255) / 256, 256, 0, stream>>>(
      x, conv1_w, conv1_b, bn1_g, bn1_b, a1);
  conv2pool_k<<<(128 * 32 * 32 * 32 + 255) / 256, 256, 0, stream>>>(
      a1, conv2_w, conv2_b, bn2_g, bn2_b, a2);
  conv3pool_k<<<(128 * 64 * 16 * 16 + 255) / 256, 256, 0, stream>>>(
      a2, conv3_w, conv3_b, bn3_g, bn3_b, a3);
  conv4_k<<<(128 * 64 * 16 * 16 + 255) / 256, 256, 0, stream>>>(
      a3, conv4_w, conv4_b, bn4_g, bn4_b, a4);
  avg8_k<<<128, 256, 0, stream>>>(a4, feat);

  // --- encoder FCs ---
  gemm_bias_wmma<<<dim3(2, 8), 128, 0, stream>>>(feat, fc1_w, fc1_b, fcm, 128, 256);
  ln_relu_k<<<128, 256, 0, stream>>>(fcm, fcln_g, fcln_b, fcm2, 128);
  gemm_bias_wmma<<<dim3(4, 8), 128, 0, stream>>>(fcm2, fc2_w, fc2_b, enc, 256, 128);
  gemm_bias_wmma<<<dim3(1, 8), 128, 0, stream>>>(enc, v_w, v_b, vals, 64, 256);

  // --- tick loop (32 serial ticks) ---
  s_init_k<<<128, 256, 0, stream>>>(enc, S);
  for (int t = 0; t < 32; ++t) {
    s_tail_k<<<256, 256, 0, stream>>>(init_post, postm, S, t);
    gemm_bias_wmma<<<dim3(4, 8), 128, 0, stream>>>(S, syn1_w, syn1_b, h1, 256, 768);
    ln_relu_k<<<128, 256, 0, stream>>>(h1, synln_g, synln_b, h1n, 256);
    gemm_bias_wmma<<<dim3(8, 8), 128, 0, stream>>>(
        h1n, syn2_w, syn2_b, pre_mem + (size_t)t * 128 * 512, 512, 256);
    pre_tr_k<<<256, 256, 0, stream>>>(pre_mem + (size_t)t * 128 * 512,
                                      preT + (size_t)t * 128 * 512);
    nlm_tick_wmma<<<512, 256, 0, stream>>>(init_pre, preT, nlm1_w, nlm1_b,
                                           nlmln_g, nlmln_b, nlm2_w, nlm2_b,
                                           postm, t);
  }

  // --- sync + readout ---
  sync_k<<<128, 256, 0, stream>>>(postm, pairs, decay, syncb);
  comb_k<<<160, 256, 0, stream>>>(vals, syncb, comb);
  gemm_bias_wmma<<<dim3(8, 8), 128, 0, stream>>>(comb, r1_w, r1_b, hr1, 512, 320);
  ln_relu_k<<<128, 256, 0, stream>>>(hr1, rln1_g, rln1_b, hr1n, 512);
  gemm_bias_wmma<<<dim3(1, 8), 128, 0, stream>>>(hr1n, r2_w, r2_b, hr2, 64, 512);
  ln_relu_k<<<128, 256, 0, stream>>>(hr2, rln2_g, rln2_b, hr2n, 64);
  gemm_bias_wmma<<<dim3(16, 8), 128, 0, stream>>>(hr2n, r3_w, r3_b, out, 1000, 64);
}